// BipartateMatching_34144990003463
// MI455X (gfx1250) — compile-verified
//
#include <hip/hip_runtime.h>
#include <hip/hip_bf16.h>

typedef __attribute__((ext_vector_type(2))) float v2f;
typedef __attribute__((ext_vector_type(8))) float v8f;

#define NPTS 4096
#define DIM  64
#define BIGF 3.4e38f

// ---------------- squared norms ----------------
__global__ void bm_norms_kernel(const float* __restrict__ X, const float* __restrict__ Y,
                                float* __restrict__ xx, float* __restrict__ yy) {
    int t = blockIdx.x * blockDim.x + threadIdx.x;
    if (t < NPTS) {
        const float* p = X + (size_t)t * DIM;
        float s = 0.f;
        #pragma unroll 8
        for (int k = 0; k < DIM; ++k) s += p[k] * p[k];
        xx[t] = s;
    } else if (t < 2 * NPTS) {
        int r = t - NPTS;
        const float* p = Y + (size_t)r * DIM;
        float s = 0.f;
        #pragma unroll 8
        for (int k = 0; k < DIM; ++k) s += p[k] * p[k];
        yy[r] = s;
    }
}

// ---------------- cost matrix via f32 WMMA ----------------
// One wave (32 lanes) computes a 16x16 tile: C = sqrt(max(xx+yy-2*X@Y^T, 0)).
// V_WMMA_F32_16X16X4_F32 layouts (ISA 7.12.2):
//   A 16x4: lanes 0-15 -> M=lane, K={0,1} in v0/v1; lanes 16-31 -> K={2,3}
//   B 4x16: mirrored (lane half selects K-pair, lane%16 = N)
//   C/D: VGPR r, lane l -> M = r + 8*(l>=16), N = l%16
__global__ void bm_cost_wmma_kernel(const float* __restrict__ X, const float* __restrict__ Y,
                                    const float* __restrict__ xx, const float* __restrict__ yy,
                                    float* __restrict__ C, float* __restrict__ CT) {
    const int tileM = blockIdx.x * 16;
    const int tileN = blockIdx.y * 16;
    const int lane  = threadIdx.x;      // 0..31, wave32
    const int half  = lane >> 4;        // 0 or 1
    const int sub   = lane & 15;

    const float* xrow = X + (size_t)(tileM + sub) * DIM;
    const float* yrow = Y + (size_t)(tileN + sub) * DIM;

    v8f acc = {};
    #pragma unroll
    for (int k = 0; k < DIM; k += 4) {
        v2f a = *(const v2f*)(xrow + k + 2 * half);
        v2f b = *(const v2f*)(yrow + k + 2 * half);
        acc = __builtin_amdgcn_wmma_f32_16x16x4_f32(false, a, false, b,
                                                    (short)0, acc, false, false);
    }

    const int n  = tileN + sub;
    const float yn = yy[n];
    #pragma unroll
    for (int r = 0; r < 8; ++r) {
        int m = tileM + r + 8 * half;
        float s = xx[m] + yn - 2.0f * acc[r];
        float d = sqrtf(fmaxf(s, 0.0f));
        C[(size_t)m * NPTS + n]  = d;
        CT[(size_t)n * NPTS + m] = d;
    }
}

// ---------------- matching state init ----------------
__global__ void bm_init_kernel(float* rowmin, int* rowarg, float* colmin, int* colarg,
                               int* rowFree, int* colFree, int* out) {
    int t = blockIdx.x * blockDim.x + threadIdx.x;
    if (t < NPTS) {
        rowmin[t] = BIGF; colmin[t] = BIGF;
        rowarg[t] = -1;   colarg[t] = -1;
        rowFree[t] = 1;   colFree[t] = 1;
        out[t] = -1;
    }
}

// ---------------- lazy per-row argmin over free cols ----------------
// One block per row. Skips work unless this row is free AND its cached
// argmin column has been matched away (free sets only shrink, so a live
// witness keeps the cached min valid).
__global__ void bm_rowscan_kernel(const float* __restrict__ M, const int* __restrict__ otherFree,
                                  const int* __restrict__ myFree,
                                  float* __restrict__ vmin, int* __restrict__ varg) {
    __shared__ float sv[256];
    __shared__ int   si[256];
    const int i = blockIdx.x;
    if (!myFree[i]) return;
    int cached = varg[i];
    if (cached >= 0 && otherFree[cached]) return;

    const float* row = M + (size_t)i * NPTS;
    float bv = BIGF; int bj = -1;
    for (int j = threadIdx.x; j < NPTS; j += 256) {
        if (otherFree[j]) {
            float c = row[j];
            if (c < bv) { bv = c; bj = j; }
        }
    }
    sv[threadIdx.x] = bv; si[threadIdx.x] = bj;
    __syncthreads();
    for (int s = 128; s > 0; s >>= 1) {
        if (threadIdx.x < s) {
            float ov = sv[threadIdx.x + s]; int oj = si[threadIdx.x + s];
            float mv = sv[threadIdx.x];     int mj = si[threadIdx.x];
            if (ov < mv || (ov == mv && (unsigned)oj < (unsigned)mj)) {
                sv[threadIdx.x] = ov; si[threadIdx.x] = oj;
            }
        }
        __syncthreads();
    }
    if (threadIdx.x == 0) { vmin[i] = sv[0]; varg[i] = si[0]; }
}

// ---------------- parallel mutual-min matching round ----------------
__global__ void bm_match_kernel(const int* __restrict__ rowarg, const int* __restrict__ colarg,
                                int* rowFree, int* colFree, int* out) {
    int j = blockIdx.x * blockDim.x + threadIdx.x;
    if (j >= NPTS) return;
    if (!colFree[j]) return;
    int i = colarg[j];
    if (i < 0) return;
    if (rowFree[i] && rowarg[i] == j) {   // mutually minimal -> greedy would pick it
        out[j] = i;
        rowFree[i] = 0;
        colFree[j] = 0;
    }
}

// ---------------- deterministic cleanup (single workgroup) ----------------
// Guarantees completion: the global-min free pair is always mutual, so each
// iteration matches >=1 pair. Normally everything is already matched and this
// exits immediately.
__global__ void bm_cleanup_kernel(const float* __restrict__ C, const float* __restrict__ CT,
                                  int* rowarg, int* colarg,
                                  int* rowFree, int* colFree, int* out) {
    __shared__ int s_cnt;
    const int tid = threadIdx.x;
    for (int iter = 0; iter < NPTS; ++iter) {
        if (tid == 0) s_cnt = 0;
        __syncthreads();
        int local = 0;
        for (int j = tid; j < NPTS; j += blockDim.x) local += colFree[j];
        if (local) atomicAdd(&s_cnt, local);
        __syncthreads();
        if (s_cnt == 0) break;

        // full rescan of free rows
        for (int i = tid; i < NPTS; i += blockDim.x) {
            if (rowFree[i]) {
                const float* row = C + (size_t)i * NPTS;
                float bv = BIGF; int bj = -1;
                for (int j = 0; j < NPTS; ++j)
                    if (colFree[j]) { float c = row[j]; if (c < bv) { bv = c; bj = j; } }
                rowarg[i] = bj;
            }
        }
        // full rescan of free cols
        for (int j = tid; j < NPTS; j += blockDim.x) {
            if (colFree[j]) {
                const float* col = CT + (size_t)j * NPTS;
                float bv = BIGF; int bi = -1;
                for (int i = 0; i < NPTS; ++i)
                    if (rowFree[i]) { float c = col[i]; if (c < bv) { bv = c; bi = i; } }
                colarg[j] = bi;
            }
        }
        __threadfence();
        __syncthreads();
        // mutual match
        for (int j = tid; j < NPTS; j += blockDim.x) {
            if (colFree[j]) {
                int i = colarg[j];
                if (i >= 0 && rowarg[i] == j) {
                    out[j] = i;
                    rowFree[i] = 0;
                    colFree[j] = 0;
                }
            }
        }
        __threadfence();
        __syncthreads();
    }
}

extern "C" void kernel_launch(void* const* d_in, const int* in_sizes, int n_in,
                              void* d_out, int out_size, void* d_ws, size_t ws_size,
                              hipStream_t stream) {
    const float* X = (const float*)d_in[0];
    const float* Y = (const float*)d_in[1];
    int* out = (int*)d_out;

    char* ws = (char*)d_ws;
    const size_t costBytes = (size_t)NPTS * NPTS * sizeof(float);   // 64 MB
    float* cost  = (float*)(ws);
    float* costT = (float*)(ws + costBytes);
    char*  small = ws + 2 * costBytes;
    float* xx     = (float*)(small);                 small += NPTS * sizeof(float);
    float* yy     = (float*)(small);                 small += NPTS * sizeof(float);
    float* rowmin = (float*)(small);                 small += NPTS * sizeof(float);
    int*   rowarg = (int*)(small);                   small += NPTS * sizeof(int);
    float* colmin = (float*)(small);                 small += NPTS * sizeof(float);
    int*   colarg = (int*)(small);                   small += NPTS * sizeof(int);
    int*   rowFree = (int*)(small);                  small += NPTS * sizeof(int);
    int*   colFree = (int*)(small);

    // 1) squared norms
    bm_norms_kernel<<<(2 * NPTS + 255) / 256, 256, 0, stream>>>(X, Y, xx, yy);

    // 2) cost matrix (and transpose) via fp32 WMMA: one wave per 16x16 tile
    dim3 gridC(NPTS / 16, NPTS / 16);
    bm_cost_wmma_kernel<<<gridC, 32, 0, stream>>>(X, Y, xx, yy, cost, costT);

    // 3) init matching state
    bm_init_kernel<<<NPTS / 256, 256, 0, stream>>>(rowmin, rowarg, colmin, colarg,
                                                   rowFree, colFree, out);

    // 4) mutual-min rounds (L2-resident scans; blocks exit once matched).
    //    Expected O(log N) ~ 25 rounds; 48 for margin, tail rounds nearly free.
    for (int r = 0; r < 48; ++r) {
        bm_rowscan_kernel<<<NPTS, 256, 0, stream>>>(cost,  colFree, rowFree, rowmin, rowarg);
        bm_rowscan_kernel<<<NPTS, 256, 0, stream>>>(costT, rowFree, colFree, colmin, colarg);
        bm_match_kernel<<<NPTS / 256, 256, 0, stream>>>(rowarg, colarg, rowFree, colFree, out);
    }

    // 5) deterministic completion guarantee
    bm_cleanup_kernel<<<1, 1024, 0, stream>>>(cost, costT, rowarg, colarg,
                                              rowFree, colFree, out);
}